// DynamicRouting_18726057410732
// MI455X (gfx1250) — compile-verified
//
#include <hip/hip_runtime.h>

typedef float v2f __attribute__((ext_vector_type(2)));
typedef float v8f __attribute__((ext_vector_type(8)));

#define B_   16
#define I_   32
#define C_   8
#define J_   10
#define D_   16
#define HW_  36
#define DHW  576            // D*H*W
#define N_   9216           // B*DHW
#define EPSF 1e-7f

// ---------------------------------------------------------------------------
// Kernel 1: channel reduction (the only HBM-heavy pass, ~94 MB read)
// P[j][i][n] = sum_c u_hat[b,i,c,j,dhw],  n = b*576 + dhw
// ---------------------------------------------------------------------------
__global__ void k_reduce(const float* __restrict__ u, float* __restrict__ P) {
    int t  = blockIdx.x * blockDim.x + threadIdx.x;   // float4 group index
    int p4 = t * 4;
    if (p4 >= I_ * J_ * N_) return;                   // 2,949,120 floats
    int ji  = p4 / N_;
    int n0  = p4 - ji * N_;
    int j   = ji / I_;
    int i   = ji - j * I_;
    int b   = n0 / DHW;
    int dhw = n0 - b * DHW;
    long base = ((long)((b * I_ + i) * C_) * J_ + j) * DHW + dhw; // c-stride = J_*DHW
    float4 acc = {0.f, 0.f, 0.f, 0.f};
#pragma unroll
    for (int c = 0; c < C_; ++c) {
        float4 vv = *reinterpret_cast<const float4*>(u + base + (long)c * (J_ * DHW));
        acc.x += vv.x; acc.y += vv.y; acc.z += vv.z; acc.w += vv.w;
    }
    *reinterpret_cast<float4*>(P + p4) = acc;
}

// ---------------------------------------------------------------------------
// Zero routing logits b_ij at the start of every call (ws is not re-poisoned)
// ---------------------------------------------------------------------------
__global__ void k_init(float* __restrict__ bij) {
    int t = threadIdx.x;
    if (t < I_ * J_) bij[t] = 0.f;
}

// ---------------------------------------------------------------------------
// Softmax over j of b_ij rows (32 rows of 10); also zeroes n1[b] accumulators
// ---------------------------------------------------------------------------
__global__ void k_softmax(const float* __restrict__ bij, float* __restrict__ cij,
                          float* __restrict__ n1) {
    int i = threadIdx.x;
    if (i < B_) n1[i] = 0.f;
    if (i < I_) {
        float row[J_];
        float m = -1e30f;
#pragma unroll
        for (int j = 0; j < J_; ++j) { row[j] = bij[i * J_ + j]; m = fmaxf(m, row[j]); }
        float s = 0.f;
#pragma unroll
        for (int j = 0; j < J_; ++j) { row[j] = __expf(row[j] - m); s += row[j]; }
        float inv = 1.f / s;
#pragma unroll
        for (int j = 0; j < J_; ++j) cij[i * J_ + j] = row[j] * inv;
    }
}

// ---------------------------------------------------------------------------
// s[j][n] = sum_i c[i][j] * P[j][i][n] + bias[j][d]    via V_WMMA_F32_16X16X4_F32
// One wave per (j, 16-n tile): A broadcasts c_j along M, B holds 16 n-columns.
// Also accumulates per-batch L1 |s| into n1[b] (wave-reduced, 1 atomic/wave).
// ---------------------------------------------------------------------------
__global__ void k_s(const float* __restrict__ P, const float* __restrict__ cij,
                    const float* __restrict__ bias, float* __restrict__ s_ws,
                    float* __restrict__ n1) {
    int lane = threadIdx.x & 31;
    int wave = blockIdx.x * (blockDim.x >> 5) + (threadIdx.x >> 5); // 0..5759
    int j    = wave / (N_ / 16);
    int nt   = wave - j * (N_ / 16);
    int nb   = nt * 16;
    int g    = lane >> 4;       // lane group -> K offset 2g
    int lm   = lane & 15;       // M (A) / N (B,C)

    v8f acc = {};
#pragma unroll
    for (int i0 = 0; i0 < I_; i0 += 4) {
        int ka = i0 + 2 * g;
        v2f a, bb;
        a.x  = cij[ka * J_ + j];                    // A[m,k]=c[i0+k][j], all m
        a.y  = cij[(ka + 1) * J_ + j];
        bb.x = P[(j * I_ + ka) * N_ + nb + lm];     // B[k,nn]=P[j][i0+k][nb+nn]
        bb.y = P[(j * I_ + ka + 1) * N_ + nb + lm];
        acc  = __builtin_amdgcn_wmma_f32_16x16x4_f32(false, a, false, bb,
                                                     (short)0, acc, false, false);
    }
    // every lane: acc[0] = D[8g, lm] = s[nb+lm] (rows identical by construction)
    float sval = acc[0];
    int n   = nb + lm;
    int b   = n / DHW;
    int dhw = n - b * DHW;
    int d   = dhw / HW_;
    sval += bias[j * D_ + d];

    float av = (lane < 16) ? fabsf(sval) : 0.f;     // count each n once
    av += __shfl_xor(av, 8, 16);
    av += __shfl_xor(av, 4, 16);
    av += __shfl_xor(av, 2, 16);
    av += __shfl_xor(av, 1, 16);
    if (lane < 16) s_ws[j * N_ + n] = sval;
    if (lane == 0) atomicAdd(&n1[b], av);           // 16-run stays within one b
}

// ---------------------------------------------------------------------------
// Squash: v = (n1^2/(1+n1^2)) * s/(n1+eps); store routing copy [j][n],
// and on the last round write d_out in reference [B,J,D,H,W] order.
// ---------------------------------------------------------------------------
__global__ void k_v(const float* __restrict__ s_ws, const float* __restrict__ n1,
                    float* __restrict__ v_ws, float* __restrict__ out, int write_out) {
    int idx = blockIdx.x * blockDim.x + threadIdx.x;
    if (idx >= J_ * N_) return;
    int j   = idx / N_;
    int n   = idx - j * N_;
    int b   = n / DHW;
    int dhw = n - b * DHW;
    float t     = n1[b];
    float t2    = t * t;
    float scale = (t2 / (1.f + t2)) * (1.f / (t + EPSF));
    float v     = scale * s_ws[idx];
    v_ws[idx] = v;
    if (write_out) out[(b * J_ + j) * DHW + dhw] = v;
}

// ---------------------------------------------------------------------------
// agreement[i,j] = sum_n P[j][i][n] * v[j][n]   via V_WMMA_F32_16X16X4_F32
// Block = (j, i-tile of 16); 8 waves stripe the 2304 K-chunks of 4 n-values.
// A holds 16 i-rows x 4 n; B broadcasts v along N; lane N==0 commits rows.
// ---------------------------------------------------------------------------
__global__ void k_agree(const float* __restrict__ P, const float* __restrict__ v_ws,
                        float* __restrict__ bij) {
    int lane = threadIdx.x & 31;
    int w    = threadIdx.x >> 5;              // 0..7
    int j    = blockIdx.x >> 1;
    int i0   = (blockIdx.x & 1) * 16;
    int g    = lane >> 4;
    int lm   = lane & 15;

    const float* Pj = P + (size_t)(j * I_ + i0 + lm) * N_;  // row i = i0+lm
    const float* vj = v_ws + (size_t)j * N_;

    v8f acc = {};
    for (int chunk = w; chunk < N_ / 4; chunk += 8) {
        int n4 = chunk * 4;
        v2f a, bb;
        a.x  = Pj[n4 + 2 * g];                // A[m,k] = P[i0+m][n4+k]
        a.y  = Pj[n4 + 2 * g + 1];
        bb.x = vj[n4 + 2 * g];                // B[k,nn] = v[n4+k] (all nn)
        bb.y = vj[n4 + 2 * g + 1];
        acc  = __builtin_amdgcn_wmma_f32_16x16x4_f32(false, a, false, bb,
                                                     (short)0, acc, false, false);
    }
    // acc[r] = D[8g+r, lm]; all 16 columns identical -> commit on lm==0 only
    if (lm == 0) {
#pragma unroll
        for (int r = 0; r < 8; ++r)
            atomicAdd(&bij[(i0 + 8 * g + r) * J_ + j], acc[r]);
    }
}

// ---------------------------------------------------------------------------
extern "C" void kernel_launch(void* const* d_in, const int* in_sizes, int n_in,
                              void* d_out, int out_size, void* d_ws, size_t ws_size,
                              hipStream_t stream) {
    const float* u    = (const float*)d_in[0];   // [B,I,C,J,D,H,W]
    const float* bias = (const float*)d_in[1];   // [J,D,1,1]
    float* out = (float*)d_out;

    float* ws   = (float*)d_ws;
    float* P    = ws;                                  // 2,949,120 floats
    float* s_ws = P + (size_t)I_ * J_ * N_;            //    92,160
    float* v_ws = s_ws + (size_t)J_ * N_;              //    92,160
    float* bij  = v_ws + (size_t)J_ * N_;              //       320
    float* cij  = bij + I_ * J_;                       //       320
    float* n1   = cij + I_ * J_;                       //        16

    // one HBM streaming pass; everything after lives in L2 (12 MB << 192 MB)
    k_reduce<<<2880, 256, 0, stream>>>(u, P);
    k_init<<<1, 512, 0, stream>>>(bij);

    for (int t = 0; t < 3; ++t) {                      // num_routing = 3
        k_softmax<<<1, 32, 0, stream>>>(bij, cij, n1);
        k_s<<<720, 256, 0, stream>>>(P, cij, bias, s_ws, n1);
        k_v<<<360, 256, 0, stream>>>(s_ws, n1, v_ws, out, t == 2 ? 1 : 0);
        if (t < 2) k_agree<<<20, 256, 0, stream>>>(P, v_ws, bij);
    }
}